// DeformableConv_37572373905604
// MI455X (gfx1250) — compile-verified
//
#include <hip/hip_runtime.h>
#include <math.h>

#define B_ 4
#define C_ 64
#define H_ 160
#define W_ 160
#define O_ 64
#define G_ 4
#define CG_ 16
#define K_ 9
#define KT_ 576          // C_*K_  reduction length of both im2col GEMMs
#define MOD_CH 36        // G_*K_  modulator channels
#define MOD_MT 3         // modulator M-tiles (36 rows padded to 48)
#define TILE_W 32        // pixels per workgroup
#define COL_STRIDE 33    // padded pixel stride in LDS (bank-conflict avoidance)
#define NKSTEP (KT_ / 4) // 144 k-steps of 4 for f32 WMMA

typedef __attribute__((ext_vector_type(2))) float v2f;
typedef __attribute__((ext_vector_type(8))) float v8f;

// sigmoid via v_exp_f32 + v_rcp_f32 (avoids the IEEE-precise div sequence)
__device__ __forceinline__ float sigmoidf_(float x) {
    return __builtin_amdgcn_rcpf(1.0f + __expf(-x));
}

// ---------------- Kernel 1: off = 80*sigmoid(offset_map) - 40 ----------------
__global__ void off_kernel(const float* __restrict__ offset_map,
                           float* __restrict__ off_out, int n) {
    int i = blockIdx.x * blockDim.x + threadIdx.x;
    if (i < n) off_out[i] = 80.0f * sigmoidf_(offset_map[i]) - 40.0f;
}

// ---------------- Kernel 2a: swizzle w_reg into WMMA A-fragment order --------
// [(mt*144 + ks)*64 + lane*2 + j] = W[mt*16 + (lane&15)][ks*4 + (lane>>4)*2 + j]
__global__ void wswz_kernel(const float* __restrict__ w_reg,
                            float* __restrict__ w_swz) {
    int idx = blockIdx.x * blockDim.x + threadIdx.x;
    if (idx >= 4 * NKSTEP * 64) return;
    int j    = idx & 1;
    int lane = (idx >> 1) & 31;
    int ks   = (idx >> 6) % NKSTEP;
    int mt   = (idx >> 6) / NKSTEP;
    int row  = mt * 16 + (lane & 15);
    int kk   = ks * 4 + (lane >> 4) * 2 + j;
    w_swz[idx] = w_reg[row * KT_ + kk];
}

// ---------------- Kernel 2b: swizzle w_mod (36x576, zero-pad to 48 rows) -----
__global__ void wmswz_kernel(const float* __restrict__ w_mod,
                             float* __restrict__ w_mswz) {
    int idx = blockIdx.x * blockDim.x + threadIdx.x;
    if (idx >= MOD_MT * NKSTEP * 64) return;
    int j    = idx & 1;
    int lane = (idx >> 1) & 31;
    int ks   = (idx >> 6) % NKSTEP;
    int mt   = (idx >> 6) / NKSTEP;
    int row  = mt * 16 + (lane & 15);
    int kk   = ks * 4 + (lane >> 4) * 2 + j;
    w_mswz[idx] = (row < MOD_CH) ? w_mod[row * KT_ + kk] : 0.0f;
}

// ---------------- Kernel 3: modulator conv via f32 WMMA ----------------------
// mod = 2*sigmoid(conv3x3(x, w_mod) + b)   as a 48x576 @ 576x32 GEMM
__global__ void __launch_bounds__(256, 1)
modulator_wmma_kernel(const float* __restrict__ x,
                      const float* __restrict__ w_mswz, // 3*144*64 swizzled
                      const float* __restrict__ b_mod,
                      float* __restrict__ mod_out) {
    __shared__ float col[KT_ * COL_STRIDE]; // ~74 KB

    const int wbase = blockIdx.x * TILE_W;
    const int h     = blockIdx.y;
    const int bb    = blockIdx.z;
    const int tid   = threadIdx.x;

    // ---- Phase 1: zero-padded 3x3 im2col into LDS (coalesced) ----
    const float* xb = x + (size_t)bb * C_ * H_ * W_;
    for (int e = tid; e < KT_ * TILE_W; e += 256) {
        int p  = e & (TILE_W - 1);
        int kt = e >> 5;            // c*9 + ky*3 + kx
        int c  = kt / K_;
        int k  = kt - c * K_;
        int ky = k / 3;
        int kx = k - 3 * ky;
        int y  = h - 1 + ky;
        int xw = wbase + p - 1 + kx;
        float v = 0.0f;
        if (y >= 0 && y < H_ && xw >= 0 && xw < W_)
            v = xb[(c * H_ + y) * W_ + xw];
        col[kt * COL_STRIDE + p] = v;
    }
    __syncthreads();

    // ---- Phase 2: 6 wave-tiles (3 M-tiles x 2 pixel tiles) ----
    const int lane = tid & 31;
    const int wave = tid >> 5;
    if (wave >= MOD_MT * 2) return; // waves 6,7 idle (wave-uniform exit)
    const int mt   = wave >> 1;     // 0..2
    const int nt   = wave & 1;      // 0..1
    const int half = lane >> 4;
    const int nn   = lane & 15;

    v8f acc = {0.f, 0.f, 0.f, 0.f, 0.f, 0.f, 0.f, 0.f};
    const v2f* wa = (const v2f*)w_mswz;

#pragma unroll 4
    for (int ks = 0; ks < NKSTEP; ++ks) {
        v2f a = wa[(mt * NKSTEP + ks) * 32 + lane];
        int kb = ks * 4 + half * 2;
        v2f bf;
        bf[0] = col[(kb + 0) * COL_STRIDE + nt * 16 + nn];
        bf[1] = col[(kb + 1) * COL_STRIDE + nt * 16 + nn];
        acc = __builtin_amdgcn_wmma_f32_16x16x4_f32(
            false, a, false, bf, (short)0, acc, false, false);
    }

    int wcol = wbase + nt * 16 + nn;
#pragma unroll
    for (int i = 0; i < 8; ++i) {
        int row = mt * 16 + half * 8 + i;
        if (row < MOD_CH) {
            float v = 2.0f * sigmoidf_(acc[i] + b_mod[row]);
            mod_out[((bb * MOD_CH + row) * H_ + h) * W_ + wcol] = v;
        }
    }
}

// ---------------- Kernel 4: deformable conv via f32 WMMA ---------------------
__global__ void __launch_bounds__(256, 1)
deform_wmma_kernel(const float* __restrict__ x,
                   const float* __restrict__ offset_map,
                   const float* __restrict__ mod,   // (B,36,H,W)
                   const float* __restrict__ w_swz, // swizzled (64 x 576)
                   float* __restrict__ out) {
    __shared__ float col[KT_ * COL_STRIDE]; // ~74 KB

    const int wbase = blockIdx.x * TILE_W;
    const int h     = blockIdx.y;
    const int bb    = blockIdx.z;
    const int tid   = threadIdx.x;

    // ---- Phase 1: masked bilinear im2col into LDS ----
    for (int t = tid; t < TILE_W * G_ * K_; t += 256) {
        int p  = t & (TILE_W - 1);
        int gk = t >> 5; // 0..35
        int g  = gk / K_;
        int k  = gk - g * K_;
        int ki = k / 3;
        int kj = k - 3 * ki;
        int w  = wbase + p;

        int offc = (g * K_ + k) * 2;
        float oy = 80.0f * sigmoidf_(offset_map[((bb * 2 * G_ * K_ + offc) * H_ + h) * W_ + w]) - 40.0f;
        float ox = 80.0f * sigmoidf_(offset_map[((bb * 2 * G_ * K_ + offc + 1) * H_ + h) * W_ + w]) - 40.0f;
        float m  = mod[((bb * MOD_CH + gk) * H_ + h) * W_ + w];

        float py = oy + (float)(h - 1 + ki);
        float px = ox + (float)(w - 1 + kj);
        float fy0 = floorf(py), fx0 = floorf(px);
        float wy = py - fy0, wx = px - fx0;
        int y0 = (int)fy0, x0 = (int)fx0;
        int y1 = y0 + 1,  x1 = x0 + 1;

        float vy0 = (y0 >= 0 && y0 < H_) ? 1.0f : 0.0f;
        float vy1 = (y1 >= 0 && y1 < H_) ? 1.0f : 0.0f;
        float vx0 = (x0 >= 0 && x0 < W_) ? 1.0f : 0.0f;
        float vx1 = (x1 >= 0 && x1 < W_) ? 1.0f : 0.0f;

        int cy0 = min(max(y0, 0), H_ - 1);
        int cy1 = min(max(y1, 0), H_ - 1);
        int cx0 = min(max(x0, 0), W_ - 1);
        int cx1 = min(max(x1, 0), W_ - 1);

        // corner weights pre-multiplied by mask and validity
        float w00 = (1.0f - wy) * (1.0f - wx) * vy0 * vx0 * m;
        float w01 = (1.0f - wy) * wx          * vy0 * vx1 * m;
        float w10 = wy          * (1.0f - wx) * vy1 * vx0 * m;
        float w11 = wy          * wx          * vy1 * vx1 * m;

        int i00 = cy0 * W_ + cx0;
        int i01 = cy0 * W_ + cx1;
        int i10 = cy1 * W_ + cx0;
        int i11 = cy1 * W_ + cx1;

        const float* xg = x + (size_t)(bb * C_ + g * CG_) * H_ * W_;
        float* dst = &col[(g * 144 + k) * COL_STRIDE + p];
#pragma unroll
        for (int c = 0; c < CG_; ++c) {
            const float* xc = xg + c * (H_ * W_);
            float v = w00 * xc[i00] + w01 * xc[i01] + w10 * xc[i10] + w11 * xc[i11];
            dst[c * K_ * COL_STRIDE] = v; // col[g*144 + c*9 + k][p]
        }
    }
    __syncthreads();

    // ---- Phase 2: 64x576 @ 576x32 GEMM, one 16x16 tile per wave32 ----
    const int lane = tid & 31;
    const int wave = tid >> 5;
    const int mt   = wave >> 1; // 0..3  output-channel tile
    const int nt   = wave & 1;  // 0..1  pixel tile
    const int half = lane >> 4;
    const int nn   = lane & 15;

    v8f acc = {0.f, 0.f, 0.f, 0.f, 0.f, 0.f, 0.f, 0.f};
    const v2f* wa = (const v2f*)w_swz;

#pragma unroll 4
    for (int ks = 0; ks < NKSTEP; ++ks) {
        v2f a = wa[(mt * NKSTEP + ks) * 32 + lane]; // coalesced b64 load
        int kb = ks * 4 + half * 2;
        v2f bf;
        bf[0] = col[(kb + 0) * COL_STRIDE + nt * 16 + nn];
        bf[1] = col[(kb + 1) * COL_STRIDE + nt * 16 + nn];
        acc = __builtin_amdgcn_wmma_f32_16x16x4_f32(
            false, a, false, bf, (short)0, acc, false, false);
    }

    // D layout: VGPR i -> rows (mt*16 + half*8 + i), col = nt*16 + nn
    float* outp = out + (((size_t)bb * O_ + mt * 16 + half * 8) * H_ + h) * W_
                      + wbase + nt * 16 + nn;
#pragma unroll
    for (int i = 0; i < 8; ++i)
        outp[(size_t)i * H_ * W_] = acc[i];
}

// ---------------- launch --------------------------------------------------
extern "C" void kernel_launch(void* const* d_in, const int* in_sizes, int n_in,
                              void* d_out, int out_size, void* d_ws, size_t ws_size,
                              hipStream_t stream) {
    const float* warp_ref   = (const float*)d_in[0];
    const float* offset_map = (const float*)d_in[1];
    const float* w_reg      = (const float*)d_in[2];
    const float* w_mod      = (const float*)d_in[3];
    const float* b_mod      = (const float*)d_in[4];

    float* out = (float*)d_out;                          // (B,O,H,W)
    float* off = out + (size_t)B_ * O_ * H_ * W_;        // (B,72,H,W)

    float* ws_mod  = (float*)d_ws;                           // (B,36,H,W)
    float* ws_wswz = ws_mod + (size_t)B_ * MOD_CH * H_ * W_; // 4*144*64 floats
    float* ws_mswz = ws_wswz + (size_t)4 * NKSTEP * 64;      // 3*144*64 floats

    int n_off = B_ * 2 * G_ * K_ * H_ * W_;
    off_kernel<<<(n_off + 255) / 256, 256, 0, stream>>>(offset_map, off, n_off);

    wswz_kernel<<<(4 * NKSTEP * 64 + 255) / 256, 256, 0, stream>>>(w_reg, ws_wswz);
    wmswz_kernel<<<(MOD_MT * NKSTEP * 64 + 255) / 256, 256, 0, stream>>>(w_mod, ws_mswz);

    dim3 grid(W_ / TILE_W, H_, B_); // (5,160,4)
    modulator_wmma_kernel<<<grid, 256, 0, stream>>>(
        warp_ref, ws_mswz, b_mod, ws_mod);

    deform_wmma_kernel<<<grid, 256, 0, stream>>>(
        warp_ref, offset_map, ws_mod, ws_wswz, out);
}